// DownSampling_7559142441750
// MI455X (gfx1250) — compile-verified
//
#include <hip/hip_runtime.h>
#include <hip/hip_bf16.h>

#ifndef __has_builtin
#define __has_builtin(x) 0
#endif

// Problem shape (fixed by the reference setup_inputs()):
#define FPS_B 16
#define FPS_N 16384
#define FPS_M (FPS_N / 4)   // 4096
#define FPS_C 128
#define TPB   1024          // 32 wave32 waves -> 8 per SIMD32, one WGP per batch
#define PPT   (FPS_N / TPB) // 16 points per thread, dist[] kept in VGPRs

typedef unsigned int u32x4 __attribute__((ext_vector_type(4)));
typedef int          i32x4 __attribute__((ext_vector_type(4)));
typedef int          i32x8 __attribute__((ext_vector_type(8)));

#if __has_builtin(__builtin_amdgcn_tensor_load_to_lds) && \
    __has_builtin(__builtin_amdgcn_s_wait_tensorcnt)
#define FPS_USE_TDM 1
#else
#define FPS_USE_TDM 0
#endif

// One workgroup per batch. Entire xyz slab (192 KB) is staged into LDS via the
// Tensor Data Mover; 320 KB WGP-LDS is what makes this layout possible on CDNA5.
__global__ __launch_bounds__(TPB) void fps_kernel(const float* __restrict__ xyz,
                                                  float* __restrict__ out_xyz,
                                                  int* __restrict__ idx_out) {
  __shared__ __attribute__((aligned(16))) float s_pts[3 * FPS_N]; // [x|y|z] planes
  __shared__ float s_wval[TPB / 32];
  __shared__ int   s_widx[TPB / 32];
  __shared__ int   s_cur;

  const int b    = blockIdx.x;
  const int t    = threadIdx.x;
  const int wave = t >> 5;
  const int lane = t & 31;
  const float* __restrict__ gx = xyz + (size_t)b * 3 * FPS_N; // contiguous [3][N]

  // ---- Stage xyz[b] (192 KB) into LDS ----
#if FPS_USE_TDM
  if (wave == 0) {
    // Tensor DMA descriptor (ISA cdna5 §8): 2-D tile, 4-byte elements,
    // tile_dim0 = 16384, tile_dim1 = 3, tensor_dim0_stride = 16384.
    unsigned long long ga = (unsigned long long)(const void*)gx;
    unsigned lds = (unsigned)(size_t)&s_pts[0]; // low 32b of flat LDS addr == LDS byte addr
    u32x4 g0;
    g0.x = 1u;                                                   // count=1, user D#
    g0.y = lds;                                                  // lds_addr (bytes)
    g0.z = (unsigned)(ga & 0xFFFFFFFFull);                       // global_addr[31:0]
    g0.w = (unsigned)((ga >> 32) & 0x01FFFFFFull) | (2u << 30);  // addr[56:32] | type=2
    i32x8 g1;
    g1[0] = (int)(2u << 16);               // workgroup_mask=0 (not in cluster), data_size=2 (4B)
    g1[1] = (int)((unsigned)FPS_N << 16);  // tensor_dim0[15:0] in bits 63:48
    g1[2] = (int)(3u << 16);               // tensor_dim0 hi=0 | tensor_dim1 lo = 3
    g1[3] = (int)((unsigned)FPS_N << 16);  // tensor_dim1 hi=0 | tile_dim0 = 16384
    g1[4] = 3;                             // tile_dim1 = 3, tile_dim2 = 0
    g1[5] = FPS_N;                         // tensor_dim0_stride (low 32) = 16384 elems
    g1[6] = 0;
    g1[7] = 0;
    i32x4 gz = {0, 0, 0, 0};               // groups 2/3 unused for a 2-D tensor
#if __clang_major__ >= 23
    i32x8 gz8 = {0, 0, 0, 0, 0, 0, 0, 0};
    __builtin_amdgcn_tensor_load_to_lds(g0, g1, gz, gz, gz8, 0);
#else
    __builtin_amdgcn_tensor_load_to_lds(g0, g1, gz, gz, 0);
#endif
    __builtin_amdgcn_s_wait_tensorcnt(0);  // TDM completion before the barrier
  }
#else
  {
    const float4* __restrict__ src = (const float4*)gx;
    float4* dst = (float4*)s_pts;
#pragma unroll
    for (int i = 0; i < (3 * FPS_N / 4) / TPB; ++i)
      dst[t + i * TPB] = src[t + i * TPB];
  }
#endif
  __syncthreads();

  // Running min-distance lives in VGPRs (16 per thread).
  float dist[PPT];
#pragma unroll
  for (int k = 0; k < PPT; ++k) dist[k] = 1e10f; // BIG sentinel, as in reference

  int cur = 0; // deterministic start at index 0 (reference semantics: emit-then-update)
  for (int m = 0; m < FPS_M; ++m) {
    if (t == 0) {
      idx_out[b * FPS_M + m] = cur;
      out_xyz[(size_t)b * 3 * FPS_M + 0 * FPS_M + m] = s_pts[cur];
      out_xyz[(size_t)b * 3 * FPS_M + 1 * FPS_M + m] = s_pts[FPS_N + cur];
      out_xyz[(size_t)b * 3 * FPS_M + 2 * FPS_M + m] = s_pts[2 * FPS_N + cur];
    }
    // Centroid broadcast: all lanes read the same LDS address (bank broadcast).
    const float cx = s_pts[cur];
    const float cy = s_pts[FPS_N + cur];
    const float cz = s_pts[2 * FPS_N + cur];

    float bestv = -1.0f;
    int   besti = 0;
#pragma unroll
    for (int k = 0; k < PPT; ++k) {
      const int j = t + k * TPB; // lanes hit 32 consecutive words -> conflict-free
      const float dx = s_pts[j] - cx;
      const float dy = s_pts[FPS_N + j] - cy;
      const float dz = s_pts[2 * FPS_N + j] - cz;
      const float d  = dx * dx + dy * dy + dz * dz;
      const float nd = fminf(dist[k], d);
      dist[k] = nd;
      if (nd > bestv) { bestv = nd; besti = j; } // ascending j => keeps lowest index on ties
    }

    // wave32 argmax tree (lowest index wins ties, matching jnp.argmax)
#pragma unroll
    for (int off = 16; off > 0; off >>= 1) {
      const float ov = __shfl_xor(bestv, off, 32);
      const int   oi = __shfl_xor(besti, off, 32);
      if (ov > bestv || (ov == bestv && oi < besti)) { bestv = ov; besti = oi; }
    }
    if (lane == 0) { s_wval[wave] = bestv; s_widx[wave] = besti; }
    __syncthreads();

    if (wave == 0) {
      float v = s_wval[lane];
      int   i = s_widx[lane];
#pragma unroll
      for (int off = 16; off > 0; off >>= 1) {
        const float ov = __shfl_xor(v, off, 32);
        const int   oi = __shfl_xor(i, off, 32);
        if (ov > v || (ov == v && oi < i)) { v = ov; i = oi; }
      }
      if (lane == 0) s_cur = i;
    }
    __syncthreads();
    cur = s_cur;
  }
}

// Feature gather: out[b,c,m] = feat[b,c,idx[b,m]]. The whole 128 MB feature
// tensor fits in the 192 MB L2, so the scattered 4B gathers stay on-chip.
__global__ void gather_kernel(const float* __restrict__ feat,
                              const int* __restrict__ idx,
                              float* __restrict__ outf) {
  const int m = blockIdx.x * blockDim.x + threadIdx.x;
  const int c = blockIdx.y;
  const int b = blockIdx.z;
  const int j = idx[b * FPS_M + m];
  outf[((size_t)b * FPS_C + c) * FPS_M + m] =
      feat[((size_t)b * FPS_C + c) * FPS_N + j];
}

extern "C" void kernel_launch(void* const* d_in, const int* in_sizes, int n_in,
                              void* d_out, int out_size, void* d_ws, size_t ws_size,
                              hipStream_t stream) {
  const float* xyz  = (const float*)d_in[0]; // [B,3,N]
  const float* feat = (const float*)d_in[1]; // [B,C,N]
  float* out_xyz  = (float*)d_out;                         // [B,3,M]
  float* out_feat = (float*)d_out + (size_t)FPS_B * 3 * FPS_M; // [B,C,M]
  int*   idx      = (int*)d_ws;                            // [B,M] scratch (256 KB)

  fps_kernel<<<dim3(FPS_B), dim3(TPB), 0, stream>>>(xyz, out_xyz, idx);
  gather_kernel<<<dim3(FPS_M / 256, FPS_C, FPS_B), dim3(256), 0, stream>>>(feat, idx, out_feat);
}